// QuickPatternMatchingLoss_3599182594549
// MI455X (gfx1250) — compile-verified
//
#include <hip/hip_runtime.h>
#include <hip/hip_bf16.h>

typedef float v2f __attribute__((ext_vector_type(2)));
typedef float v8f __attribute__((ext_vector_type(8)));

#define LSZ 512   // sequence length (== padded size)
#define NCH 21    // x channels
#define HPROF 30  // seq_hmm channels
#define WAVES_PER_BLOCK 8

// One wave == one batch sample. Per 16-position group:
//   logits(3x16) = Wpad(16x20) * src(20x16) + g(3x16)   via 5 chained fp32 WMMAs
// with the batch-invariant seq_hmm contribution g precomputed into LDS per block.
__global__ __launch_bounds__(256) void qpml_kernel(
    const float* __restrict__ x,      // [B,21,512]
    const float* __restrict__ seq,    // [30,512]
    const float* __restrict__ ss,     // [3,512]
    const float* __restrict__ W,      // [3,50]
    const float* __restrict__ bias,   // [3]
    float* __restrict__ out,          // [B]
    int B)
{
    __shared__ float g_s[3 * LSZ];   // g[o,l] = b[o] + sum_h W[o,20+h]*seq[h,l]
    __shared__ float w_s[3 * LSZ];   // ss_hmm copy

    const int tid = threadIdx.x;

    // ---- Phase 1: batch-invariant tables into LDS (block-cooperative) ----
    for (int i = tid; i < 3 * LSZ; i += 256) {
        const int o = i >> 9;          // i / 512
        const int l = i & (LSZ - 1);
        float acc = bias[o];
        const float* wr = W + o * 50 + 20;
        #pragma unroll
        for (int h = 0; h < HPROF; ++h)
            acc = fmaf(wr[h], seq[h * LSZ + l], acc);
        g_s[i] = acc;
        w_s[i] = ss[i];
    }
    __syncthreads();

    const int wave = tid >> 5;
    const int lane = tid & 31;
    const int batch = blockIdx.x * WAVES_PER_BLOCK + wave;
    if (batch >= B) return;            // uniform per wave

    const float* xb = x + (size_t)batch * NCH * LSZ;
    const int m  = lane & 15;          // A row (output o) / B column (position)
    const int hi = lane >> 4;          // K-half select per fp32 WMMA layout

    // A tiles: W rows 0..2 (x-channel part, src channels 0..19), rows 3..15 zero.
    // fp32 16x4 A layout: lane m holds row m; VGPR0/1 = K {2*hi, 2*hi+1} of chunk.
    v2f A[5];
    #pragma unroll
    for (int j = 0; j < 5; ++j) {
        const int c0 = 4 * j + 2 * hi;                  // src channel < 20
        A[j].x = (m < 3) ? W[m * 50 + c0]     : 0.0f;
        A[j].y = (m < 3) ? W[m * 50 + c0 + 1] : 0.0f;
    }

    float acc = 0.0f, cnt = 0.0f;

    for (int gq = 0; gq < 32; ++gq) {
        const int pos = (gq << 4) + m;                  // this lane's position

        // B tiles: src channels 0..19 == x channels 1..20; each x value loaded
        // exactly once across the lane pair (hi splits K). 4x16 B layout:
        // lanes 0-15 rows {k,k+1}, lanes 16-31 rows {k+2,k+3}.
        v2f Bv[5];
        #pragma unroll
        for (int j = 0; j < 5; ++j) {
            const int xc = 4 * j + 2 * hi + 1;          // x channel index
            Bv[j].x = xb[xc * LSZ + pos];
            Bv[j].y = xb[(xc + 1) * LSZ + pos];
        }

        // Argmax over all 21 x channels, reusing the B loads.
        float bestv = -3.0e38f;
        int   besti = 0;
        if (hi == 0) { bestv = xb[pos]; besti = 0; }    // channel 0 (gap channel)
        #pragma unroll
        for (int j = 0; j < 5; ++j) {
            const int c = 4 * j + 2 * hi + 1;           // increasing order -> first-max wins
            if (Bv[j].x > bestv) { bestv = Bv[j].x; besti = c; }
            if (Bv[j].y > bestv) { bestv = Bv[j].y; besti = c + 1; }
        }
        {   // combine lane pair (channels interleave; tie -> smaller index)
            const float ov = __shfl_xor(bestv, 16, 32);
            const int   oi = __shfl_xor(besti, 16, 32);
            if (ov > bestv || (ov == bestv && oi < besti)) { bestv = ov; besti = oi; }
        }
        const float maskf = (besti != 0) ? 1.0f : 0.0f;

        // C = g[o, pos] in rows 0..2 (D rows 3..15 are dead: A rows are zero).
        v8f d;
        d[0] = g_s[pos];
        d[1] = g_s[LSZ + pos];
        d[2] = g_s[2 * LSZ + pos];
        d[3] = 0.0f; d[4] = 0.0f; d[5] = 0.0f; d[6] = 0.0f; d[7] = 0.0f;

        d = __builtin_amdgcn_wmma_f32_16x16x4_f32(false, A[0], false, Bv[0], (short)0, d, false, false);
        d = __builtin_amdgcn_wmma_f32_16x16x4_f32(false, A[1], false, Bv[1], (short)0, d, false, false);
        d = __builtin_amdgcn_wmma_f32_16x16x4_f32(false, A[2], false, Bv[2], (short)0, d, false, false);
        d = __builtin_amdgcn_wmma_f32_16x16x4_f32(false, A[3], false, Bv[3], (short)0, d, false, false);
        d = __builtin_amdgcn_wmma_f32_16x16x4_f32(false, A[4], false, Bv[4], (short)0, d, false, false);

        // Lanes 0-15: D[0..2] = logits o=0..2 for position `pos`.
        const float l0 = d[0], l1 = d[1], l2 = d[2];
        const float mx = fmaxf(l0, fmaxf(l1, l2));
        const float e0 = __expf(l0 - mx);
        const float e1 = __expf(l1 - mx);
        const float e2 = __expf(l2 - mx);
        const float num = w_s[pos] * e0 + w_s[LSZ + pos] * e1 + w_s[2 * LSZ + pos] * e2;
        // den in [1,3]: single v_rcp_f32 instead of IEEE divide sequence.
        const float p   = num * __builtin_amdgcn_rcpf(e0 + e1 + e2);

        const float take = (hi == 0) ? maskf : 0.0f;    // high lanes hold dead rows
        acc = fmaf(take, p, acc);
        cnt += take;
    }

    // Wave reduction (high lanes contribute zeros).
    #pragma unroll
    for (int off = 16; off > 0; off >>= 1) {
        acc += __shfl_xor(acc, off, 32);
        cnt += __shfl_xor(cnt, off, 32);
    }
    if (lane == 0)
        out[batch] = __logf(acc / cnt);
}

extern "C" void kernel_launch(void* const* d_in, const int* in_sizes, int n_in,
                              void* d_out, int out_size, void* d_ws, size_t ws_size,
                              hipStream_t stream) {
    const float* x    = (const float*)d_in[0];   // [B,21,512]
    const float* seq  = (const float*)d_in[1];   // [30,512]
    const float* ss   = (const float*)d_in[2];   // [3,512]
    const float* W    = (const float*)d_in[3];   // [3,50]
    const float* bias = (const float*)d_in[4];   // [3]
    float* out = (float*)d_out;

    const int B = in_sizes[0] / (NCH * LSZ);     // 4096
    const int grid = (B + WAVES_PER_BLOCK - 1) / WAVES_PER_BLOCK;
    qpml_kernel<<<grid, 256, 0, stream>>>(x, seq, ss, W, bias, out, B);
}